// RoIPool_40862318854289
// MI455X (gfx1250) — compile-verified
//
#include <hip/hip_runtime.h>
#include <stdint.h>

// Problem constants (fixed by reference setup_inputs)
#define PH_ 7
#define PW_ 7
#define SCALE_ 0.0625f
#define NB_ 4
#define C_ 128
#define H_ 50
#define W_ 50
#define K_ 256
#define CG_ 4   // channels staged per block

typedef __attribute__((address_space(3))) int lds_int_t;
typedef __attribute__((address_space(1))) int glob_int_t;

__device__ __forceinline__ void async_load_f32_to_lds(const float* gptr, float* lptr) {
#if __has_builtin(__builtin_amdgcn_global_load_async_to_lds_b32)
    __builtin_amdgcn_global_load_async_to_lds_b32(
        (glob_int_t*)gptr, (lds_int_t*)lptr, 0, 0);
#else
    unsigned loff = (unsigned)(uintptr_t)(lds_int_t*)lptr;       // LDS byte address
    unsigned long long ga = (unsigned long long)(uintptr_t)gptr; // 64-bit global addr
    asm volatile("global_load_async_to_lds_b32 %0, %1, off"
                 :: "v"(loff), "v"(ga) : "memory");
#endif
}

__device__ __forceinline__ void wait_async0() {
#if __has_builtin(__builtin_amdgcn_s_wait_asynccnt)
    __builtin_amdgcn_s_wait_asynccnt(0);
#else
    asm volatile("s_wait_asynccnt 0" ::: "memory");
#endif
}

__global__ __launch_bounds__(256)
void RoIPool_40862318854289_kernel(const float* __restrict__ feat,
                                   const float* __restrict__ rois,
                                   float* __restrict__ out)
{
    __shared__ float buf[CG_ * H_ * W_];   // 40 KB worst-case region for 4 channels

    const int k   = blockIdx.y;          // ROI index
    const int cb  = blockIdx.x * CG_;    // channel-group base
    const int tid = threadIdx.x;

    // --- ROI geometry (redundant per thread; matches reference quantization) ---
    const float* r = rois + k * 5;
    const int b  = (int)r[0];
    const int x1 = (int)rintf(r[1] * SCALE_);   // jnp.round = round-half-even
    const int y1 = (int)rintf(r[2] * SCALE_);
    const int x2 = (int)rintf(r[3] * SCALE_);
    const int y2 = (int)rintf(r[4] * SCALE_);
    const float roi_w = (float)max(x2 - x1 + 1, 1);
    const float roi_h = (float)max(y2 - y1 + 1, 1);
    const float bin_w = roi_w / (float)PW_;
    const float bin_h = roi_h / (float)PH_;

    // Staging region = union of all bin windows: [rh0,rh1) x [rw0,rw1)
    const int rh0 = min(max(y1, 0), H_);
    const int rh1 = min(max((int)ceilf(7.0f * bin_h) + y1, 0), H_);  // == hend[6]
    const int rw0 = min(max(x1, 0), W_);
    const int rw1 = min(max((int)ceilf(7.0f * bin_w) + x1, 0), W_);  // == wend[6]
    const int regH = rh1 - rh0;
    const int regW = rw1 - rw0;
    const int RHW  = regH * regW;
    const int E    = CG_ * RHW;

    // --- Phase 1: async-stage the region for CG_ channels into LDS ---
    for (int idx = tid; idx < E; idx += 256) {
        int g  = idx / RHW;
        int rm = idx - g * RHW;
        int h  = rm / regW;
        int w  = rm - h * regW;
        const float* gp = feat + (((b * C_ + cb + g) * H_ + (rh0 + h)) * W_ + (rw0 + w));
        async_load_f32_to_lds(gp, &buf[idx]);
    }
    wait_async0();      // each wave drains its own ASYNCcnt (LDS writes landed)
    __syncthreads();    // publish across all 8 waves of the workgroup

    // --- Phase 2: 4 channels x 49 bins = 196 outputs per block ---
    if (tid < CG_ * PH_ * PW_) {
        int g   = tid / (PH_ * PW_);
        int bin = tid - g * (PH_ * PW_);
        int ph  = bin / PW_;
        int pw  = bin - ph * PW_;

        int hs = min(max((int)floorf((float)ph * bin_h) + y1, 0), H_);
        int he = min(max((int)ceilf(((float)ph + 1.0f) * bin_h) + y1, 0), H_);
        int ws = min(max((int)floorf((float)pw * bin_w) + x1, 0), W_);
        int we = min(max((int)ceilf(((float)pw + 1.0f) * bin_w) + x1, 0), W_);

        float v = 0.0f;                      // empty-bin semantics
        if (he > hs && we > ws) {
            float m = -1e30f;                // reference NEG sentinel
            const float* base = &buf[g * RHW];
            for (int h = hs; h < he; ++h) {
                const float* row = base + (h - rh0) * regW - rw0;
                for (int w = ws; w < we; ++w) m = fmaxf(m, row[w]);
            }
            v = m;
        }
        out[((k * C_ + cb + g) * PH_ + ph) * PW_ + pw] = v;
    }
}

extern "C" void kernel_launch(void* const* d_in, const int* in_sizes, int n_in,
                              void* d_out, int out_size, void* d_ws, size_t ws_size,
                              hipStream_t stream) {
    (void)in_sizes; (void)n_in; (void)d_ws; (void)ws_size; (void)out_size;
    const float* feat = (const float*)d_in[0];   // [4,128,50,50] f32
    const float* rois = (const float*)d_in[1];   // [256,5] f32
    float* out = (float*)d_out;                  // [256,128,7,7] f32
    dim3 grid(C_ / CG_, K_);                     // (32, 256) blocks
    hipLaunchKernelGGL(RoIPool_40862318854289_kernel, grid, dim3(256), 0, stream,
                       feat, rois, out);
}